// GELU213_23648089932117
// MI455X (gfx1250) — compile-verified
//
#include <hip/hip_runtime.h>
#include <math.h>

#define GEPS      1e-5f
#define GEPS_VAR  1e-4f
#define GNORM_EPS 1e-12f

#if defined(__HIP_DEVICE_COMPILE__)
#  if __has_builtin(__builtin_amdgcn_wmma_f32_16x16x4_f32)
#    define HAVE_WMMA4 1
#  endif
#  if __has_builtin(__builtin_amdgcn_global_load_async_to_lds_b128) && \
      __has_builtin(__builtin_amdgcn_s_wait_asynccnt)
#    define HAVE_ASYNC 1
#  endif
#endif

typedef __attribute__((ext_vector_type(2))) float v2f;
typedef __attribute__((ext_vector_type(4))) float v4f;
typedef __attribute__((ext_vector_type(8))) float v8f;
typedef __attribute__((ext_vector_type(4))) int   v4i;

__device__ __forceinline__ float gelu_tanh(float v) {
    const float c = 0.7978845608028654f; // sqrt(2/pi)
    float u = c * (v + 0.044715f * v * v * v);
    return 0.5f * v * (1.0f + tanhf(u));
}

// ---------------------------------------------------------------------------
// Pass 1: per-channel sum|z| partials, per-row sum(out^2) and dot(out, ema_dir)
// One block = 64 consecutive rows. Channel c in LDS accum owned by exactly one
// thread across all rows -> deterministic accumulation, no atomics.
// ---------------------------------------------------------------------------
__global__ void __launch_bounds__(256) gelu213_pass1(
    const float* __restrict__ x,
    const float* __restrict__ ema_mean,
    const float* __restrict__ ema_sq,
    const float* __restrict__ ema_dir,
    float* __restrict__ partials,   // [gridDim.x * D]
    float* __restrict__ row_s2,     // [nrows]
    float* __restrict__ row_sd,     // [nrows]
    int D, int rows_per_block)
{
    __shared__ float accum[2048];
    __shared__ float red_s2[8];
    __shared__ float red_sd[8];
    const int tid  = threadIdx.x;
    const int lane = tid & 31;
    const int wv   = tid >> 5;

    for (int c = tid; c < D; c += 256) accum[c] = 0.0f;
    __syncthreads();

    const int nvec = D >> 2;
    const size_t row0 = (size_t)blockIdx.x * rows_per_block;
    const v4f* mean4 = (const v4f*)ema_mean;
    const v4f* sq4   = (const v4f*)ema_sq;
    const v4f* dir4  = (const v4f*)ema_dir;

    for (int r = 0; r < rows_per_block; ++r) {
        const size_t row = row0 + r;
        const v4f* xr = (const v4f*)(x + row * (size_t)D);
        float s2 = 0.0f, sd = 0.0f;
        for (int v = tid; v < nvec; v += 256) {
            v4f xv = xr[v];
            v4f mv = mean4[v];
            v4f qv = sq4[v];
            v4f dv = dir4[v];
            const int c = v << 2;
#pragma unroll
            for (int k = 0; k < 4; ++k) {
                float o   = gelu_tanh(xv[k]);
                float var = fmaxf(qv[k] - mv[k] * mv[k], GEPS_VAR);
                float z   = (o - mv[k]) / (sqrtf(var) + GEPS);
                accum[c + k] += fabsf(z);
                s2 += o * o;
                sd += o * dv[k];
            }
        }
#pragma unroll
        for (int off = 16; off > 0; off >>= 1) {
            s2 += __shfl_xor(s2, off, 32);
            sd += __shfl_xor(sd, off, 32);
        }
        if (lane == 0) { red_s2[wv] = s2; red_sd[wv] = sd; }
        __syncthreads();
        if (tid == 0) {
            float a = 0.0f, b = 0.0f;
#pragma unroll
            for (int w = 0; w < 8; ++w) { a += red_s2[w]; b += red_sd[w]; }
            row_s2[row] = a;
            row_sd[row] = b;
        }
        __syncthreads();
    }
    float* pb = partials + (size_t)blockIdx.x * D;
    for (int c = tid; c < D; c += 256) pb[c] = accum[c];
}

// ---------------------------------------------------------------------------
// Pass 2 (single block): reduce partials over blocks via WMMA (ones x P),
// top-16 mask, inv_s, ||ema_dir||, per-row cosine gate.
// ---------------------------------------------------------------------------
__global__ void __launch_bounds__(256) gelu213_pass2(
    const float* __restrict__ ema_mean,
    const float* __restrict__ ema_sq,
    const float* __restrict__ ema_dir,
    const float* __restrict__ log_tau,
    const float* __restrict__ partials, int nparts,
    const float* __restrict__ row_s2,
    const float* __restrict__ row_sd,
    float* __restrict__ chansum,    // [D]
    float* __restrict__ mask,       // [D]
    float* __restrict__ inv_s,      // [D]
    float* __restrict__ gcos,       // [nrows]
    int D, int nrows)
{
    __shared__ float red[8];
    __shared__ float nrm_sh;
    const int tid  = threadIdx.x;
    const int lane = tid & 31;
    const int wv   = tid >> 5;

    // --- (a) channel sums: chansum[c] = sum_g partials[g][c] ---
#if defined(HAVE_WMMA4)
    {
        // D = ones(16x4) x P-chunk(4x16) accumulated over K; with all-ones A,
        // every D row equals the column sums regardless of B's k-slot order.
        const int ngroups = D >> 4;
        const v2f a_ones = {1.0f, 1.0f};
        const int j  = lane & 15;
        const int kh = lane >> 4;
        for (int g = wv; g < ngroups; g += 8) {   // uniform per wave; EXEC all 1s
            const int c0 = g << 4;
            v8f acc = {0.f, 0.f, 0.f, 0.f, 0.f, 0.f, 0.f, 0.f};
            for (int k0 = 0; k0 < nparts; k0 += 4) {
                v2f b;
                b[0] = partials[(size_t)(k0 + 2 * kh    ) * D + c0 + j];
                b[1] = partials[(size_t)(k0 + 2 * kh + 1) * D + c0 + j];
                acc = __builtin_amdgcn_wmma_f32_16x16x4_f32(
                    false, a_ones, false, b, (short)0, acc, false, false);
            }
            if (lane < 16) chansum[c0 + j] = acc[0];  // (M=0, N=j)
        }
    }
#else
    for (int c = tid; c < D; c += 256) {
        float s = 0.0f;
        for (int g = 0; g < nparts; ++g) s += partials[(size_t)g * D + c];
        chansum[c] = s;
    }
#endif

    // --- (b) init mask, precompute inv_s ---
    for (int c = tid; c < D; c += 256) {
        mask[c] = 0.0f;
        float m   = ema_mean[c];
        float var = fmaxf(ema_sq[c] - m * m, GEPS_VAR);
        inv_s[c]  = 1.0f / (sqrtf(var) + GEPS);
    }

    // --- (c) ||ema_dir|| ---
    float ss = 0.0f;
    for (int c = tid; c < D; c += 256) { float d = ema_dir[c]; ss += d * d; }
#pragma unroll
    for (int off = 16; off > 0; off >>= 1) ss += __shfl_xor(ss, off, 32);
    if (lane == 0) red[wv] = ss;
    __syncthreads();
    if (tid == 0) {
        float t = 0.0f;
#pragma unroll
        for (int w = 0; w < 8; ++w) t += red[w];
        nrm_sh = sqrtf(t);
    }
    __syncthreads();

    // --- (d) deterministic top-16 (first index wins ties, like jax top_k) ---
    if (tid == 0) {
        for (int it = 0; it < 16; ++it) {
            float best = -1.0f; int bi = 0;   // sums of |z| are >= 0
            for (int c = 0; c < D; ++c) {
                float v = chansum[c];
                if (v > best) { best = v; bi = c; }
            }
            mask[bi]    = 1.0f;
            chansum[bi] = -2.0f;
        }
    }

    // --- (e) per-row cosine gate ---
    const float tau = expf(log_tau[0]);
    const float en  = fmaxf(nrm_sh, GNORM_EPS);
    for (int r = tid; r < nrows; r += 256) {
        float no = fmaxf(sqrtf(row_s2[r]), GNORM_EPS);
        float cv = row_sd[r] / (no * en);
        cv = fminf(fmaxf(cv, -1.0f), 1.0f);
        gcos[r] = expf(-tau * cv);
    }
}

// ---------------------------------------------------------------------------
// Pass 3: stream x -> LDS (double-buffered async b128) -> gated output.
// Tile = 2 rows (4096 f32 = 16 KiB); each of 256 threads issues 4 x b128.
// x should still be L2-resident from pass 1 (128 MiB < 192 MB L2); output
// stores are non-temporal to avoid evicting it.
// ---------------------------------------------------------------------------
#if defined(HAVE_ASYNC)
// Diagnostic from round 1 showed arg0 is 'int4 __device__*' i.e. addrspace(1)
// pointer to a 4 x i32 vector; LDS side is the addrspace(3) counterpart.
typedef v4i __attribute__((address_space(1)))* gas_v4i;
typedef v4i __attribute__((address_space(3)))* las_v4i;
__device__ __forceinline__ void async_ld_b128(const float* g, float* l) {
    __builtin_amdgcn_global_load_async_to_lds_b128((gas_v4i)g, (las_v4i)l, 0, 0);
}
#endif

__global__ void __launch_bounds__(256) gelu213_pass3(
    const float* __restrict__ x,
    const float* __restrict__ ema_mean,
    const float* __restrict__ inv_s,
    const float* __restrict__ mask,
    const float* __restrict__ gcos,
    const float* __restrict__ log_beta_up,
    const float* __restrict__ log_beta_dn,
    const float* __restrict__ log_gamma,
    float* __restrict__ out,
    int D, int rows_per_block)
{
    __shared__ float buf[2][4096];
    const int tid = threadIdx.x;

    const float beta_up = log1pf(expf(log_beta_up[0]));
    const float beta_dn = log1pf(expf(log_beta_dn[0]));
    const float gamma   = log1pf(expf(log_gamma[0]));

    const size_t row0   = (size_t)blockIdx.x * rows_per_block;
    const int tile_rows = 2;
    const int ntiles    = rows_per_block / tile_rows;

    // prologue: issue tile 0
    {
        const float* src = x + row0 * (size_t)D;
        float* dst = buf[0];
#pragma unroll
        for (int i = 0; i < 4; ++i) {
            const int v = tid + (i << 8);
#if defined(HAVE_ASYNC)
            async_ld_b128(src + (v << 2), dst + (v << 2));
#else
            ((v4f*)dst)[v] = ((const v4f*)src)[v];
#endif
        }
    }

    for (int t = 0; t < ntiles; ++t) {
        if (t + 1 < ntiles) {
            const float* src = x + (row0 + (size_t)(t + 1) * tile_rows) * D;
            float* dst = buf[(t + 1) & 1];
#pragma unroll
            for (int i = 0; i < 4; ++i) {
                const int v = tid + (i << 8);
#if defined(HAVE_ASYNC)
                async_ld_b128(src + (v << 2), dst + (v << 2));
#else
                ((v4f*)dst)[v] = ((const v4f*)src)[v];
#endif
            }
#if defined(HAVE_ASYNC)
            __builtin_amdgcn_s_wait_asynccnt(4);   // current tile done, next in flight
#endif
        } else {
#if defined(HAVE_ASYNC)
            __builtin_amdgcn_s_wait_asynccnt(0);
#endif
        }
        __syncthreads();

        const float* bp  = buf[t & 1];
        const size_t trow = row0 + (size_t)t * tile_rows;
        const float g0 = gcos[trow];
        const float g1 = gcos[trow + 1];
#pragma unroll
        for (int i = 0; i < 4; ++i) {
            const int v  = tid + (i << 8);
            const int e  = v << 2;
            const int rl = (e >= D) ? 1 : 0;
            const int c  = e - rl * D;
            const float gc = rl ? g1 : g0;
            v4f xv = ((const v4f*)bp)[v];
            v4f mv = ((const v4f*)ema_mean)[c >> 2];
            v4f iv = ((const v4f*)inv_s)[c >> 2];
            v4f kv = ((const v4f*)mask)[c >> 2];
            v4f res;
#pragma unroll
            for (int k = 0; k < 4; ++k) {
                float o    = gelu_tanh(xv[k]);
                float z    = (o - mv[k]) * iv[k];
                float th   = tanhf(gamma * z);
                float graw = 1.0f + beta_up * fmaxf(th, 0.0f) - beta_dn * fmaxf(-th, 0.0f);
                float gate = (kv[k] != 0.0f) ? fminf(fmaxf(graw, 0.05f), 8.0f) : 1.0f;
                res[k] = o * gate * gc;
            }
            __builtin_nontemporal_store(res, (v4f*)(out + trow * (size_t)D) + v);
        }
        __syncthreads();
    }
}

// ---------------------------------------------------------------------------
extern "C" void kernel_launch(void* const* d_in, const int* in_sizes, int n_in,
                              void* d_out, int out_size, void* d_ws, size_t ws_size,
                              hipStream_t stream) {
    (void)n_in; (void)out_size; (void)ws_size;
    const float* x            = (const float*)d_in[0];
    const float* log_tau      = (const float*)d_in[1];
    const float* log_beta_up  = (const float*)d_in[2];
    const float* log_beta_dn  = (const float*)d_in[3];
    const float* log_gamma    = (const float*)d_in[4];
    const float* ema_mean     = (const float*)d_in[5];
    const float* ema_sq       = (const float*)d_in[6];
    const float* ema_dir      = (const float*)d_in[7];

    const int D     = in_sizes[5];        // 2048
    const int nrows = in_sizes[0] / D;    // 16384

    const int G1   = 256;
    const int rpb1 = nrows / G1;          // 64

    float* ws       = (float*)d_ws;
    float* partials = ws;                              // G1*D
    float* row_s2   = partials + (size_t)G1 * D;       // nrows
    float* row_sd   = row_s2 + nrows;                  // nrows
    float* chansum  = row_sd + nrows;                  // D
    float* maskv    = chansum + D;                     // D
    float* inv_s    = maskv + D;                       // D
    float* gcosv    = inv_s + D;                       // nrows

    gelu213_pass1<<<G1, 256, 0, stream>>>(x, ema_mean, ema_sq, ema_dir,
                                          partials, row_s2, row_sd, D, rpb1);
    gelu213_pass2<<<1, 256, 0, stream>>>(ema_mean, ema_sq, ema_dir, log_tau,
                                         partials, G1, row_s2, row_sd,
                                         chansum, maskv, inv_s, gcosv, D, nrows);
    const int G3   = 1024;
    const int rpb3 = nrows / G3;          // 16 rows -> 8 tiles of 2 rows
    gelu213_pass3<<<G3, 256, 0, stream>>>(x, ema_mean, inv_s, maskv, gcosv,
                                          log_beta_up, log_beta_dn, log_gamma,
                                          (float*)d_out, D, rpb3);
}